// InteractionBlock_61753039782727
// MI455X (gfx1250) — compile-verified
//
#include <hip/hip_runtime.h>
#include <hip/hip_bf16.h>

// Problem sizes (fixed by the reference)
#define NN 20000
#define EE 320000

// 1/sqrt(d) constants from the reference (_inv is rsqrt!)
#define INV_C    0.25f                  // 1/sqrt(16)
#define INV_3C   0.14433756729740643f   // 1/sqrt(48)
#define INV_3    0.57735026918962580f   // 1/sqrt(3)
#define RC2      0.70710678118654760f   // 1/sqrt(2)
#define RC6      0.40824829046386310f   // 1/sqrt(6)
#define NORM_AVG 0.25f                  // 1/sqrt(AVG=16)

typedef __attribute__((ext_vector_type(2))) float v2f;
typedef __attribute__((ext_vector_type(8))) float v8f;

__device__ __forceinline__ float silu_f(float x) {
    return x / (1.0f + __expf(-x));
}

// ---------------------------------------------------------------------------
// Kernel 1: node up-projections  s = x@W_up_s * 1/sqrt(C);  v likewise.
// ---------------------------------------------------------------------------
__global__ __launch_bounds__(256) void up_kernel(
    const float* __restrict__ xs, const float* __restrict__ xv,
    const float* __restrict__ Ws, const float* __restrict__ Wv,
    float* __restrict__ sUp, float* __restrict__ vUp)
{
    int idx = blockIdx.x * 256 + threadIdx.x;
    if (idx >= NN * 16) return;
    int n = idx >> 4, d = idx & 15;
    float acc = 0.0f, a0 = 0.0f, a1 = 0.0f, a2 = 0.0f;
#pragma unroll
    for (int c = 0; c < 16; ++c) {
        float w_s = Ws[c * 16 + d];
        float w_v = Wv[c * 16 + d];
        acc += xs[n * 16 + c] * w_s;
        a0  += xv[(n * 16 + c) * 3 + 0] * w_v;
        a1  += xv[(n * 16 + c) * 3 + 1] * w_v;
        a2  += xv[(n * 16 + c) * 3 + 2] * w_v;
    }
    sUp[idx]         = acc * INV_C;
    vUp[idx * 3 + 0] = a0 * INV_C;
    vUp[idx * 3 + 1] = a1 * INV_C;
    vUp[idx * 3 + 2] = a2 * INV_C;
}

// ---------------------------------------------------------------------------
// Kernel 2: fused edge kernel.
//   Per wave: 16 edges. WMMA f32 16x16x4 radial MLP:
//     H[16x64]  = silu(emb[16x8] @ W_rad1[8x64])      (8 WMMAs)
//     Wt[16x112]= H @ W_rad2[64x112]                  (112 WMMAs)
//   Then per-(edge,channel) tensor products + atomic scatter into ns/nv/n2.
//   WMMA register layouts (ISA 7.12.2):
//     A frag (16x4):  elem(j,lane) = A[lane&15][kbase + j + 2*(lane>>4)]
//     B frag (4x16):  elem(j,lane) = B[kbase + j + 2*(lane>>4)][nbase + (lane&15)]
//     C frag (16x16): elem(r,lane) = C[r + 8*(lane>>4)][nbase + (lane&15)]
// ---------------------------------------------------------------------------
__global__ __launch_bounds__(128) void edge_kernel(
    const float* __restrict__ edge_Y0, const float* __restrict__ edge_Y1,
    const float* __restrict__ emb, const float* __restrict__ W_rad1,
    const float* __restrict__ W_rad2, const int* __restrict__ senders,
    const int* __restrict__ receivers, const float* __restrict__ sUp,
    const float* __restrict__ vUp, float* __restrict__ NSb,
    float* __restrict__ NVb, float* __restrict__ N2b)
{
    __shared__ float ldsH[4][16 * 64];   // per-wave H tile (edge-major)
    __shared__ float ldsW[64 * 112];     // per-block edge weights (norm folded)
    __shared__ float sY0[64];
    __shared__ float sY1[64][3];
    __shared__ int   sSnd[64];
    __shared__ int   sRcv[64];

    const int tid  = threadIdx.x;
    const int wave = tid >> 5;
    const int lane = tid & 31;
    const int m    = lane & 15;
    const int hi   = lane >> 4;
    const int e0   = (blockIdx.x * 4 + wave) * 16;

    // Stage per-edge scalars (one lane per edge)
    if (lane < 16) {
        int e  = e0 + lane;
        int el = wave * 16 + lane;
        sY0[el]    = edge_Y0[e];
        sY1[el][0] = edge_Y1[e * 3 + 0];
        sY1[el][1] = edge_Y1[e * 3 + 1];
        sY1[el][2] = edge_Y1[e * 3 + 2];
        sSnd[el]   = senders[e];
        sRcv[el]   = receivers[e];
    }

    // ---- GEMM1: H = silu(emb @ W_rad1) ----
    v8f Hc[4];
#pragma unroll
    for (int n = 0; n < 4; ++n)
#pragma unroll
        for (int r = 0; r < 8; ++r) Hc[n][r] = 0.0f;

#pragma unroll
    for (int ks = 0; ks < 2; ++ks) {
        v2f a;
        a[0] = emb[(e0 + m) * 8 + ks * 4 + 0 + 2 * hi];
        a[1] = emb[(e0 + m) * 8 + ks * 4 + 1 + 2 * hi];
#pragma unroll
        for (int n = 0; n < 4; ++n) {
            v2f b;
            b[0] = W_rad1[(ks * 4 + 0 + 2 * hi) * 64 + n * 16 + m];
            b[1] = W_rad1[(ks * 4 + 1 + 2 * hi) * 64 + n * 16 + m];
            Hc[n] = __builtin_amdgcn_wmma_f32_16x16x4_f32(
                false, a, false, b, (short)0, Hc[n], false, false);
        }
    }
    // silu + store H tile to LDS: row = edge (M), col = hidden unit (N)
#pragma unroll
    for (int n = 0; n < 4; ++n)
#pragma unroll
        for (int r = 0; r < 8; ++r)
            ldsH[wave][(r + 8 * hi) * 64 + n * 16 + m] = silu_f(Hc[n][r]);

    __syncthreads();

    // Reload H as A-fragments for GEMM2 (K = hidden dim)
    v2f aF[16];
#pragma unroll
    for (int ks = 0; ks < 16; ++ks) {
        aF[ks][0] = ldsH[wave][m * 64 + ks * 4 + 0 + 2 * hi];
        aF[ks][1] = ldsH[wave][m * 64 + ks * 4 + 1 + 2 * hi];
    }

    // ---- GEMM2: Wt = H @ W_rad2, 7 N-tiles of 16 ----
#pragma unroll 1
    for (int nt = 0; nt < 7; ++nt) {
        v8f Cc;
#pragma unroll
        for (int r = 0; r < 8; ++r) Cc[r] = 0.0f;
#pragma unroll
        for (int ks = 0; ks < 16; ++ks) {
            v2f b;
            b[0] = W_rad2[(ks * 4 + 0 + 2 * hi) * 112 + nt * 16 + m];
            b[1] = W_rad2[(ks * 4 + 1 + 2 * hi) * 112 + nt * 16 + m];
            Cc = __builtin_amdgcn_wmma_f32_16x16x4_f32(
                false, aF[ks], false, b, (short)0, Cc, false, false);
        }
        // Store weights (norm 1/sqrt(AVG) folded in), indexed [edge][chan]
#pragma unroll
        for (int r = 0; r < 8; ++r)
            ldsW[(wave * 16 + r + 8 * hi) * 112 + nt * 16 + m] = Cc[r] * NORM_AVG;
    }

    __syncthreads();

    // ---- Phase 2: tensor products + atomic scatter (64 edges x 16 chans) ----
    for (int p = tid; p < 64 * 16; p += 128) {
        const int el  = p >> 4;
        const int c   = p & 15;
        const int snd = sSnd[el];
        const int rcv = sRcv[el];
        const float y0 = sY0[el];
        const float bx = sY1[el][0], by = sY1[el][1], bz = sY1[el][2];
        const float se = sUp[snd * 16 + c];
        const float ax = vUp[(snd * 16 + c) * 3 + 0];
        const float ay = vUp[(snd * 16 + c) * 3 + 1];
        const float az = vUp[(snd * 16 + c) * 3 + 2];
        const float* w = &ldsW[el * 112];

        // scalar messages -> ns[rcv, 0:48]
        atomicAdd(&NSb[rcv * 48 + c],      se * w[c]);
        atomicAdd(&NSb[rcv * 48 + 16 + c], se * y0 * w[16 + c]);
        float dotv = (ax * bx + ay * by + az * bz) * INV_3;
        atomicAdd(&NSb[rcv * 48 + 32 + c], dotv * w[32 + c]);

        // vector messages -> nv[rcv, 0:48, 3]
        float we = w[48 + c], ws1 = w[64 + c], wv0 = w[80 + c];
        float* nvp = &NVb[rcv * 144];
        atomicAdd(&nvp[c * 3 + 0], ax * we);
        atomicAdd(&nvp[c * 3 + 1], ay * we);
        atomicAdd(&nvp[c * 3 + 2], az * we);
        atomicAdd(&nvp[(16 + c) * 3 + 0], se * bx * ws1);
        atomicAdd(&nvp[(16 + c) * 3 + 1], se * by * ws1);
        atomicAdd(&nvp[(16 + c) * 3 + 2], se * bz * ws1);
        atomicAdd(&nvp[(32 + c) * 3 + 0], ax * y0 * wv0);
        atomicAdd(&nvp[(32 + c) * 3 + 1], ay * y0 * wv0);
        atomicAdd(&nvp[(32 + c) * 3 + 2], az * y0 * wv0);

        // L=2 messages -> n2[rcv, c, 0:5]
        float w2 = w[96 + c];
        float* n2p = &N2b[(rcv * 16 + c) * 5];
        atomicAdd(&n2p[0], (ax * by + ay * bx) * RC2 * w2);
        atomicAdd(&n2p[1], (ay * bz + az * by) * RC2 * w2);
        atomicAdd(&n2p[2], (2.0f * az * bz - ax * bx - ay * by) * RC6 * w2);
        atomicAdd(&n2p[3], (ax * bz + az * bx) * RC2 * w2);
        atomicAdd(&n2p[4], (ax * bx - ay * by) * RC2 * w2);
    }
}

// ---------------------------------------------------------------------------
// Kernel 3: per-node down-projection + species self-connection + gating.
// One block (96 threads) per node; everything staged in LDS.
// ---------------------------------------------------------------------------
__global__ __launch_bounds__(96) void node_kernel(
    const float* __restrict__ NSb, const float* __restrict__ NVb,
    const float* __restrict__ N2b, const float* __restrict__ Wds,
    const float* __restrict__ Wdv, const float* __restrict__ Wd2,
    const float* __restrict__ Wscs, const float* __restrict__ Wscv,
    const float* __restrict__ Wsc2, const int* __restrict__ species,
    float* __restrict__ out)
{
    __shared__ float sNS[48], sNV[144], sN2[80];
    __shared__ float sFS[48], sFV[48], sF2[80];
    __shared__ float sFS2[48], sFV2[48], sF22[80];
    const int n  = blockIdx.x;
    const int t  = threadIdx.x;
    const int sp = species[n];

    for (int i = t; i < 272; i += 96) {
        if (i < 48)       sNS[i]       = NSb[n * 48  + i];
        else if (i < 192) sNV[i - 48]  = NVb[n * 144 + (i - 48)];
        else              sN2[i - 192] = N2b[n * 80  + (i - 192)];
    }
    __syncthreads();

    if (t < 48) {                      // fs = ns @ W_down_s * 1/sqrt(48)
        float acc = 0.0f;
        for (int c = 0; c < 48; ++c) acc += sNS[c] * Wds[c * 48 + t];
        sFS[t] = acc * INV_3C;
    } else {                           // fv[d][x]
        int idx = t - 48, d = idx / 3, x = idx % 3;
        float acc = 0.0f;
        for (int c = 0; c < 48; ++c) acc += sNV[c * 3 + x] * Wdv[c * 16 + d];
        sFV[idx] = acc * INV_3C;
    }
    if (t < 80) {                      // f2[d][m]
        int d = t / 5, mm = t % 5;
        float acc = 0.0f;
        for (int c = 0; c < 16; ++c) acc += sN2[c * 5 + mm] * Wd2[c * 16 + d];
        sF2[t] = acc * INV_C;
    }
    __syncthreads();

    if (t < 48) {                      // fs += fs @ W_sc_s[sp] * 1/sqrt(48)
        float acc = 0.0f;
        const float* W = &Wscs[sp * 48 * 48];
        for (int c = 0; c < 48; ++c) acc += sFS[c] * W[c * 48 + t];
        sFS2[t] = sFS[t] + acc * INV_3C;
    } else {                           // fv += fv @ W_sc_v[sp] * 1/sqrt(16)
        int idx = t - 48, d = idx / 3, x = idx % 3;
        float acc = 0.0f;
        const float* W = &Wscv[sp * 256];
        for (int c = 0; c < 16; ++c) acc += sFV[c * 3 + x] * W[c * 16 + d];
        sFV2[idx] = sFV[idx] + acc * INV_C;
    }
    if (t < 80) {                      // f2 += f2 @ W_sc_2[sp] * 1/sqrt(16)
        int d = t / 5, mm = t % 5;
        float acc = 0.0f;
        const float* W = &Wsc2[sp * 256];
        for (int c = 0; c < 16; ++c) acc += sF2[c * 5 + mm] * W[c * 16 + d];
        sF22[t] = sF2[t] + acc * INV_C;
    }
    __syncthreads();

    if (t < 16) {
        out[n * 16 + t] = silu_f(sFS2[t]);
    } else if (t < 64) {
        int idx = t - 16, d = idx / 3;
        out[NN * 16 + n * 48 + idx] = sFV2[idx] * silu_f(sFS2[16 + d]);
    }
    if (t < 80) {
        out[NN * 64 + n * 80 + t] = sF22[t] * silu_f(sFS2[32 + t / 5]);
    }
}

// ---------------------------------------------------------------------------
extern "C" void kernel_launch(void* const* d_in, const int* in_sizes, int n_in,
                              void* d_out, int out_size, void* d_ws, size_t ws_size,
                              hipStream_t stream) {
    const float* node_scalars = (const float*)d_in[0];
    const float* node_vectors = (const float*)d_in[1];
    const float* edge_Y0      = (const float*)d_in[2];
    const float* edge_Y1      = (const float*)d_in[3];
    const float* edge_len_emb = (const float*)d_in[4];
    const float* W_up_s       = (const float*)d_in[5];
    const float* W_up_v       = (const float*)d_in[6];
    const float* W_rad1       = (const float*)d_in[7];
    const float* W_rad2       = (const float*)d_in[8];
    const float* W_down_s     = (const float*)d_in[9];
    const float* W_down_v     = (const float*)d_in[10];
    const float* W_down_2     = (const float*)d_in[11];
    const float* W_sc_s       = (const float*)d_in[12];
    const float* W_sc_v       = (const float*)d_in[13];
    const float* W_sc_2       = (const float*)d_in[14];
    const int*   senders      = (const int*)d_in[15];
    const int*   receivers    = (const int*)d_in[16];
    const int*   species      = (const int*)d_in[17];
    float* out = (float*)d_out;
    float* ws  = (float*)d_ws;

    // Workspace layout (floats): sUp[N*16] | vUp[N*48] | ns[N*48] | nv[N*144] | n2[N*80]
    float* sUp = ws;
    float* vUp = ws + (size_t)NN * 16;
    float* NSb = ws + (size_t)NN * 64;
    float* NVb = ws + (size_t)NN * 112;
    float* N2b = ws + (size_t)NN * 256;

    // Zero the atomic accumulators every call (graph-capture safe)
    hipMemsetAsync(NSb, 0, (size_t)NN * 272 * sizeof(float), stream);

    up_kernel<<<(NN * 16 + 255) / 256, 256, 0, stream>>>(
        node_scalars, node_vectors, W_up_s, W_up_v, sUp, vUp);

    edge_kernel<<<EE / 64, 128, 0, stream>>>(
        edge_Y0, edge_Y1, edge_len_emb, W_rad1, W_rad2,
        senders, receivers, sUp, vUp, NSb, NVb, N2b);

    node_kernel<<<NN, 96, 0, stream>>>(
        NSb, NVb, N2b, W_down_s, W_down_v, W_down_2,
        W_sc_s, W_sc_v, W_sc_2, species, out);
}